// BERT_89017492177573
// MI455X (gfx1250) — compile-verified
//
#include <hip/hip_runtime.h>
#include <hip/hip_bf16.h>
#include <math.h>

typedef __attribute__((ext_vector_type(2))) float v2f;
typedef __attribute__((ext_vector_type(8))) float v8f;

#define WMMA_F32(a, b, c) \
    __builtin_amdgcn_wmma_f32_16x16x4_f32(false, (a), false, (b), (short)0, (c), false, false)

// ---------------- model dims (compile-time constants) ----------------
static constexpr int Bd  = 8;
static constexpr int Ld  = 512;
static constexpr int Dd  = 768;
static constexpr int Hd  = 12;
static constexpr int DH  = 64;     // Dd / Hd
static constexpr int FFd = 3072;
static constexpr int NLd = 6;
static constexpr int NCd = 1;
static constexpr int Md  = Bd * Ld;   // 4096 rows

// =====================================================================
// Embedding + (buggy per-batch) positional encoding
//   h[b,l,d] = emb[x[b,l],d] + pe[b,d]
//   pe[b,2i]=sin(b*div_i), pe[b,2i+1]=cos(b*div_i)
// =====================================================================
__global__ __launch_bounds__(256) void embed_kernel(const int* __restrict__ x,
                                                    const float* __restrict__ emb,
                                                    float* __restrict__ h) {
    int idx = blockIdx.x * 256 + threadIdx.x;          // < B*L*D
    int d   = idx % Dd;
    int bl  = idx / Dd;
    int b   = bl / Ld;
    int tok = x[bl];
    int i2  = (d >> 1) * 2;
    float div = expf((float)i2 * (-9.210340371976184f / (float)Dd)); // -ln(1e4)/D
    float ang = (float)b * div;
    float pe  = (d & 1) ? cosf(ang) : sinf(ang);
    h[idx] = emb[(size_t)tok * Dd + d] + pe;
}

// =====================================================================
// LayerNorm: one block per row of D=768, 256 threads, 3 elems/thread
// =====================================================================
__global__ __launch_bounds__(256) void ln_kernel(const float* __restrict__ X,
                                                 const float* __restrict__ g,
                                                 const float* __restrict__ be,
                                                 float* __restrict__ Y) {
    __shared__ float s1[256], s2[256];
    int row = blockIdx.x, tid = threadIdx.x;
    const float* xr = X + (size_t)row * Dd;
    float a0 = xr[tid], a1 = xr[tid + 256], a2 = xr[tid + 512];
    s1[tid] = a0 + a1 + a2;
    s2[tid] = a0 * a0 + a1 * a1 + a2 * a2;
    __syncthreads();
    for (int st = 128; st > 0; st >>= 1) {
        if (tid < st) { s1[tid] += s1[tid + st]; s2[tid] += s2[tid + st]; }
        __syncthreads();
    }
    float mean = s1[0] * (1.0f / Dd);
    float var  = s2[0] * (1.0f / Dd) - mean * mean;
    float inv  = rsqrtf(var + 1e-5f);
    float* yr = Y + (size_t)row * Dd;
    yr[tid]       = (a0 - mean) * inv * g[tid]       + be[tid];
    yr[tid + 256] = (a1 - mean) * inv * g[tid + 256] + be[tid + 256];
    yr[tid + 512] = (a2 - mean) * inv * g[tid + 512] + be[tid + 512];
}

// =====================================================================
// Tiled WMMA GEMM: C[M,N] = act(A[M,K] @ Bw[K,N] + bias) (+ R)
// 64x64 block tile, BK=16, 8 waves; each wave: one 16-row tile x two
// 16-col tiles via V_WMMA_F32_16X16X4_F32.
// =====================================================================
template <bool GELU_ACT, bool RESID>
__global__ __launch_bounds__(256) void gemm_kernel(const float* __restrict__ A,
                                                   const float* __restrict__ Bw,
                                                   const float* __restrict__ bias,
                                                   const float* __restrict__ R,
                                                   float* __restrict__ C,
                                                   int M, int N, int K) {
    __shared__ float As[64][17];   // padded: stride 17 kills bank conflicts
    __shared__ float Bs[16][68];

    int tid = threadIdx.x, lane = tid & 31, w = tid >> 5;
    int bm = blockIdx.y * 64, bn = blockIdx.x * 64;
    int mt = w & 3, nt0 = (w >> 2) * 2;
    int kshift = (lane < 16) ? 0 : 2;
    int ml = mt * 16 + (lane & 15);
    int nl1 = nt0 * 16 + (lane & 15);
    int nl2 = nl1 + 16;

    v8f acc0 = {0, 0, 0, 0, 0, 0, 0, 0};
    v8f acc1 = {0, 0, 0, 0, 0, 0, 0, 0};

    int arow = tid >> 2, ac4 = (tid & 3) * 4;    // A: 64 rows x 16 cols
    int brow = tid >> 4, bc4 = (tid & 15) * 4;   // B: 16 rows x 64 cols

    for (int k0 = 0; k0 < K; k0 += 16) {
        float4 av = *(const float4*)(A  + (size_t)(bm + arow) * K + k0 + ac4);
        float4 bv = *(const float4*)(Bw + (size_t)(k0 + brow) * N + bn + bc4);
        As[arow][ac4 + 0] = av.x; As[arow][ac4 + 1] = av.y;
        As[arow][ac4 + 2] = av.z; As[arow][ac4 + 3] = av.w;
        Bs[brow][bc4 + 0] = bv.x; Bs[brow][bc4 + 1] = bv.y;
        Bs[brow][bc4 + 2] = bv.z; Bs[brow][bc4 + 3] = bv.w;
        __syncthreads();
#pragma unroll
        for (int kk = 0; kk < 4; ++kk) {
            int kb = kk * 4 + kshift;
            v2f a  = {As[ml][kb], As[ml][kb + 1]};
            v2f b0 = {Bs[kb][nl1], Bs[kb + 1][nl1]};
            acc0 = WMMA_F32(a, b0, acc0);
            v2f b1 = {Bs[kb][nl2], Bs[kb + 1][nl2]};
            acc1 = WMMA_F32(a, b1, acc1);
        }
        __syncthreads();
    }

    int nc1 = bn + nl1, nc2 = bn + nl2;
    float bia1 = bias[nc1], bia2 = bias[nc2];
#pragma unroll
    for (int rr = 0; rr < 8; ++rr) {
        int m = bm + mt * 16 + rr + ((lane < 16) ? 0 : 8);
        float v1 = acc0[rr] + bia1;
        float v2 = acc1[rr] + bia2;
        if (GELU_ACT) {
            v1 = 0.5f * v1 * (1.0f + erff(v1 * 0.7071067811865475f));
            v2 = 0.5f * v2 * (1.0f + erff(v2 * 0.7071067811865475f));
        }
        if (RESID) {
            v1 += R[(size_t)m * N + nc1];
            v2 += R[(size_t)m * N + nc2];
        }
        C[(size_t)m * N + nc1] = v1;
        C[(size_t)m * N + nc2] = v2;
    }
}

// =====================================================================
// Fused attention per (b, head, 16 query rows):
//   phase 1: S = Q·K^T / scale + bias + mask   (WMMA, 8 waves x 4 tiles)
//   phase 2: row softmax over 512 keys in LDS; stream P to attn output
//   phase 3: O = P·V (WMMA, 4 waves x one 16-col tile, full K=512)
// =====================================================================
__global__ __launch_bounds__(256) void attn_kernel(const float* __restrict__ Qm,
                                                   const float* __restrict__ Km,
                                                   const float* __restrict__ Vm,
                                                   const int* __restrict__ mask,
                                                   float* __restrict__ attn_l,
                                                   float* __restrict__ O,
                                                   float invScale) {
    __shared__ float S[16][516];     // 16 q-rows x 512 keys (padded)
    __shared__ float red[16][16];
    __shared__ float rowmax[16];
    __shared__ float rowsum[16];

    int qt = blockIdx.x, hd = blockIdx.y, b = blockIdx.z;
    int tid = threadIdx.x, lane = tid & 31, w = tid >> 5;
    int q0 = qt * 16;
    int hcol = hd * DH;
    int kshift = (lane < 16) ? 0 : 2;
    int mrow = q0 + (lane & 15);
    const float* Qrow = Qm + (size_t)(b * Ld + mrow) * Dd + hcol;

    // ---- phase 1: scores ----
    for (int i = 0; i < 4; ++i) {
        int n0   = (w * 4 + i) * 16;
        int ncol = n0 + (lane & 15);
        const float* Krow = Km + (size_t)(b * Ld + ncol) * Dd + hcol;
        v8f acc = {0, 0, 0, 0, 0, 0, 0, 0};
#pragma unroll
        for (int kk = 0; kk < 16; ++kk) {
            int kb = kk * 4 + kshift;
            v2f a  = {Qrow[kb], Qrow[kb + 1]};
            v2f bb = {Krow[kb], Krow[kb + 1]};
            acc = WMMA_F32(a, bb, acc);
        }
        int km = mask[b * Ld + ncol];
#pragma unroll
        for (int r = 0; r < 8; ++r) {
            int m = r + ((lane < 16) ? 0 : 8);
            float val = acc[r] * invScale - 0.01f * ((float)(q0 + m) - (float)ncol);
            if (km == 0) val = -__builtin_inff();
            S[m][ncol] = val;
        }
    }
    __syncthreads();

    // ---- phase 2: softmax (16 rows x 16 threads/row x 32 cols/thread) ----
    int r = tid >> 4, j = tid & 15, c0 = j * 32;
    float mx = -__builtin_inff();
    for (int c = c0; c < c0 + 32; ++c) mx = fmaxf(mx, S[r][c]);
    red[r][j] = mx;
    __syncthreads();
    if (j == 0) {
        float m2 = -__builtin_inff();
        for (int t = 0; t < 16; ++t) m2 = fmaxf(m2, red[r][t]);
        rowmax[r] = m2;
    }
    __syncthreads();
    float rm = rowmax[r], sm = 0.f;
    for (int c = c0; c < c0 + 32; ++c) {
        float e = expf(S[r][c] - rm);
        S[r][c] = e;
        sm += e;
    }
    red[r][j] = sm;
    __syncthreads();
    if (j == 0) {
        float s2 = 0.f;
        for (int t = 0; t < 16; ++t) s2 += red[r][t];
        rowsum[r] = s2;
    }
    __syncthreads();
    float inv = 1.0f / rowsum[r];
    size_t abase = (((size_t)(b * Hd + hd)) * Ld + (q0 + r)) * Ld;
    for (int c = c0; c < c0 + 32; ++c) {
        float p = S[r][c] * inv;
        S[r][c] = p;
        attn_l[abase + c] = p;
    }
    __syncthreads();

    // ---- phase 3: O = P @ V ----
    if (w < 4) {
        int n0   = w * 16;
        int ncol = n0 + (lane & 15);
        int mlq  = lane & 15;
        v8f acc = {0, 0, 0, 0, 0, 0, 0, 0};
        for (int kc = 0; kc < 128; ++kc) {
            int kb = kc * 4 + kshift;
            v2f a = {S[mlq][kb], S[mlq][kb + 1]};
            const float* Vrow = Vm + (size_t)(b * Ld + kb) * Dd + hcol + ncol;
            v2f bb = {Vrow[0], Vrow[Dd]};
            acc = WMMA_F32(a, bb, acc);
        }
#pragma unroll
        for (int rr = 0; rr < 8; ++rr) {
            int m = rr + ((lane < 16) ? 0 : 8);
            O[(size_t)(b * Ld + q0 + m) * Dd + hcol + ncol] = acc[rr];
        }
    }
}

// =====================================================================
// Classifier: logits[b,c] = h[b,0,:] . Wc[:,c] + bc[c]
// =====================================================================
__global__ __launch_bounds__(256) void cls_kernel(const float* __restrict__ Hb,
                                                  const float* __restrict__ Wc,
                                                  const float* __restrict__ bc,
                                                  float* __restrict__ outp) {
    __shared__ float s[256];
    int bcix = blockIdx.x;                 // b*NC + c
    int b = bcix / NCd, c = bcix % NCd;
    int tid = threadIdx.x;
    const float* hr = Hb + (size_t)(b * Ld) * Dd;   // CLS token (l=0)
    float acc = hr[tid]       * Wc[(tid)       * NCd + c]
              + hr[tid + 256] * Wc[(tid + 256) * NCd + c]
              + hr[tid + 512] * Wc[(tid + 512) * NCd + c];
    s[tid] = acc;
    __syncthreads();
    for (int st = 128; st > 0; st >>= 1) {
        if (tid < st) s[tid] += s[tid + st];
        __syncthreads();
    }
    if (tid == 0) outp[bcix] = s[0] + bc[c];
}

// =====================================================================
extern "C" void kernel_launch(void* const* d_in, const int* in_sizes, int n_in,
                              void* d_out, int out_size, void* d_ws, size_t ws_size,
                              hipStream_t stream) {
    const int*   x    = (const int*)d_in[0];
    const int*   mask = (const int*)d_in[1];
    const float* emb  = (const float*)d_in[2];
    const float* Wq   = (const float*)d_in[3];
    const float* bq   = (const float*)d_in[4];
    const float* Wk   = (const float*)d_in[5];
    const float* bk   = (const float*)d_in[6];
    const float* Wv   = (const float*)d_in[7];
    const float* bv   = (const float*)d_in[8];
    const float* Wo   = (const float*)d_in[9];
    const float* bo   = (const float*)d_in[10];
    const float* W1   = (const float*)d_in[11];
    const float* b1   = (const float*)d_in[12];
    const float* W2   = (const float*)d_in[13];
    const float* b2   = (const float*)d_in[14];
    const float* g1   = (const float*)d_in[15];
    const float* be1  = (const float*)d_in[16];
    const float* g2   = (const float*)d_in[17];
    const float* be2  = (const float*)d_in[18];
    const float* Wc   = (const float*)d_in[19];
    const float* bc   = (const float*)d_in[20];

    float* out   = (float*)d_out;
    float* attnb = out + (size_t)Bd * NCd;     // attn_all follows logits

    // workspace layout
    float* ws = (float*)d_ws;
    const size_t SLOT = (size_t)Md * Dd;       // 4096*768
    float* h   = ws;
    float* xn  = ws + 1 * SLOT;
    float* Qb  = ws + 2 * SLOT;
    float* Kb  = ws + 3 * SLOT;
    float* Vb  = ws + 4 * SLOT;
    float* Ob  = ws + 5 * SLOT;
    float* FFh = ws + 6 * SLOT;                // 4096*3072

    const float invScale = 1.0f / sqrtf((float)Dd / (float)Hd);  // 1/8

    embed_kernel<<<(Md * Dd) / 256, 256, 0, stream>>>(x, emb, h);

    dim3 gProj(Dd / 64, Md / 64);      // (12, 64)
    dim3 gFF1(FFd / 64, Md / 64);      // (48, 64)
    dim3 gAttn(Ld / 16, Hd, Bd);       // (32, 12, 8)

    for (int l = 0; l < NLd; ++l) {
        const float* WqL = Wq + (size_t)l * Dd * Dd;
        const float* WkL = Wk + (size_t)l * Dd * Dd;
        const float* WvL = Wv + (size_t)l * Dd * Dd;
        const float* WoL = Wo + (size_t)l * Dd * Dd;
        const float* W1L = W1 + (size_t)l * Dd * FFd;
        const float* W2L = W2 + (size_t)l * FFd * Dd;
        const float* bqL = bq + (size_t)l * Dd;
        const float* bkL = bk + (size_t)l * Dd;
        const float* bvL = bv + (size_t)l * Dd;
        const float* boL = bo + (size_t)l * Dd;
        const float* b1L = b1 + (size_t)l * FFd;
        const float* b2L = b2 + (size_t)l * Dd;
        const float* g1L = g1 + (size_t)l * Dd;
        const float* be1L = be1 + (size_t)l * Dd;
        const float* g2L = g2 + (size_t)l * Dd;
        const float* be2L = be2 + (size_t)l * Dd;
        float* attnL = attnb + (size_t)l * Bd * Hd * Ld * Ld;

        ln_kernel<<<Md, 256, 0, stream>>>(h, g1L, be1L, xn);

        gemm_kernel<false, false><<<gProj, 256, 0, stream>>>(xn, WqL, bqL, nullptr, Qb, Md, Dd, Dd);
        gemm_kernel<false, false><<<gProj, 256, 0, stream>>>(xn, WkL, bkL, nullptr, Kb, Md, Dd, Dd);
        gemm_kernel<false, false><<<gProj, 256, 0, stream>>>(xn, WvL, bvL, nullptr, Vb, Md, Dd, Dd);

        attn_kernel<<<gAttn, 256, 0, stream>>>(Qb, Kb, Vb, mask, attnL, Ob, invScale);

        // h = h + (O @ Wo + bo)
        gemm_kernel<false, true><<<gProj, 256, 0, stream>>>(Ob, WoL, boL, h, h, Md, Dd, Dd);

        ln_kernel<<<Md, 256, 0, stream>>>(h, g2L, be2L, xn);

        // FFh = gelu(xn @ W1 + b1)
        gemm_kernel<true, false><<<gFF1, 256, 0, stream>>>(xn, W1L, b1L, nullptr, FFh, Md, FFd, Dd);
        // h = h + (FFh @ W2 + b2)
        gemm_kernel<false, true><<<gProj, 256, 0, stream>>>(FFh, W2L, b2L, h, h, Md, Dd, FFd);
    }

    cls_kernel<<<Bd * NCd, 256, 0, stream>>>(h, Wc, bc, out);
}